// ExitRouter_26362509263282
// MI455X (gfx1250) — compile-verified
//
#include <hip/hip_runtime.h>

typedef float v2f __attribute__((ext_vector_type(2)));
typedef float v8f __attribute__((ext_vector_type(8)));

#define B_LEN   4
#define T_LEN   8192
#define D_LEN   2048
#define KSEL    4096           // max(1, min(T, int(0.5*T+0.5)))
#define NROWS   (B_LEN * T_LEN)

// ---------------------------------------------------------------------------
// Kernel 1: scores = sigmoid(h @ W^T + b) using V_WMMA_F32_16X16X4_F32.
// One wave per 16-row tile. A = 16x4 chunk of h rows; B = W chunk broadcast
// to all 16 columns, so every column of D holds the row dot products.
// ---------------------------------------------------------------------------
__global__ __launch_bounds__(256) void gemv_sigmoid_wmma(
    const float* __restrict__ h, const float* __restrict__ W,
    const float* __restrict__ bias, float* __restrict__ scores)
{
    __shared__ float wsh[D_LEN];           // 8 KB: W staged once per block
    const int tid = threadIdx.x;

    #pragma unroll
    for (int i = 0; i < D_LEN / 256; ++i)
        wsh[tid + i * 256] = W[tid + i * 256];
    __syncthreads();

    const int lane = tid & 31;
    const int wave = tid >> 5;                       // 0..7
    const int rowbase = (blockIdx.x * 8 + wave) * 16; // 2048 tiles exactly

    // A-matrix layout (16x4 f32): lane L -> row (L&15), K offset (L>>4)*2
    const int m    = lane & 15;
    const int koff = (lane >> 4) << 1;               // 0 or 2
    const float* aptr = h + (size_t)(rowbase + m) * D_LEN + koff;
    const float* bptr = wsh + koff;

    v8f c0 = {}, c1 = {}, c2 = {}, c3 = {};
    for (int kb = 0; kb < D_LEN; kb += 16) {
        v2f a0 = *(const v2f*)(aptr + kb + 0);
        v2f a1 = *(const v2f*)(aptr + kb + 4);
        v2f a2 = *(const v2f*)(aptr + kb + 8);
        v2f a3 = *(const v2f*)(aptr + kb + 12);
        v2f b0 = *(const v2f*)(bptr + kb + 0);
        v2f b1 = *(const v2f*)(bptr + kb + 4);
        v2f b2 = *(const v2f*)(bptr + kb + 8);
        v2f b3 = *(const v2f*)(bptr + kb + 12);
        c0 = __builtin_amdgcn_wmma_f32_16x16x4_f32(false, a0, false, b0, (short)0, c0, false, false);
        c1 = __builtin_amdgcn_wmma_f32_16x16x4_f32(false, a1, false, b1, (short)0, c1, false, false);
        c2 = __builtin_amdgcn_wmma_f32_16x16x4_f32(false, a2, false, b2, (short)0, c2, false, false);
        c3 = __builtin_amdgcn_wmma_f32_16x16x4_f32(false, a3, false, b3, (short)0, c3, false, false);
    }
    v8f cs = (c0 + c1) + (c2 + c3);

    // D layout: VGPR j, lanes 0-15 hold M=j; lanes 16-31 hold M=j+8.
    // Lanes 0-7 emit rows 0..7, lanes 16-23 emit rows 8..15.
    const int j = lane & 7;
    float val = 0.0f;
    #pragma unroll
    for (int e = 0; e < 8; ++e)
        if (j == e) val = cs[e];

    if ((lane & 15) < 8) {
        const int row = rowbase + j + ((lane >> 4) << 3);
        const float logit = val + bias[0];
        scores[row] = 1.0f / (1.0f + __expf(-logit));
    }
}

// ---------------------------------------------------------------------------
// Kernel 2: per-batch exact top-k (k = T/2) via 32-step binary search on the
// positive-float bit patterns, then mask = topk & (score > 0.5) & ~exited.
// One 1024-thread block per batch; scores live in LDS.
// ---------------------------------------------------------------------------
__global__ __launch_bounds__(1024) void topk_mask_kernel(
    const float* __restrict__ scores, const unsigned char* __restrict__ exited,
    float* __restrict__ mask)
{
    __shared__ unsigned int sc[T_LEN];     // 32 KB
    __shared__ int cnt;

    const int b   = blockIdx.x;
    const int tid = threadIdx.x;
    const float* srow = scores + (size_t)b * T_LEN;

    unsigned int v[8];
    #pragma unroll
    for (int i = 0; i < 8; ++i) {
        const int idx = tid * 8 + i;
        const unsigned int u = __float_as_uint(srow[idx]);  // sigmoid > 0 => monotone bits
        v[i] = u;
        sc[idx] = u;
    }
    __syncthreads();

    // Binary search MSB->LSB for thr = k-th largest bit pattern:
    // largest u with |{x : x >= u}| >= k.
    unsigned int thr = 0u;
    for (int bit = 31; bit >= 0; --bit) {
        const unsigned int cand = thr | (1u << bit);
        if (tid == 0) cnt = 0;
        __syncthreads();
        int c = 0;
        #pragma unroll
        for (int i = 0; i < 8; ++i) c += (v[i] >= cand) ? 1 : 0;
        for (int off = 16; off; off >>= 1) c += __shfl_xor(c, off, 32);
        if ((tid & 31) == 0) atomicAdd(&cnt, c);
        __syncthreads();
        if (cnt >= KSEL) thr = cand;
        __syncthreads();                   // protect cnt before next reset
    }

    // G = strictly-greater count; E = #threshold-valued elements to admit.
    if (tid == 0) cnt = 0;
    __syncthreads();
    {
        int c = 0;
        #pragma unroll
        for (int i = 0; i < 8; ++i) c += (v[i] > thr) ? 1 : 0;
        for (int off = 16; off; off >>= 1) c += __shfl_xor(c, off, 32);
        if ((tid & 31) == 0) atomicAdd(&cnt, c);
    }
    __syncthreads();
    const int E = KSEL - cnt;
    __syncthreads();

    #pragma unroll
    for (int i = 0; i < 8; ++i) {
        const int idx = tid * 8 + i;
        const unsigned int u = v[i];
        bool sel = false;
        if (u > thr) {
            sel = true;
        } else if (u == thr) {
            int r = 0;                     // rank among equals, lowest index first
            for (int q = 0; q < idx; ++q) r += (sc[q] == thr) ? 1 : 0;
            sel = (r < E);
        }
        const float f = __uint_as_float(u);
        const bool ex = exited[(size_t)b * T_LEN + idx] != 0;
        mask[(size_t)b * T_LEN + idx] = (sel && (f > 0.5f) && !ex) ? 1.0f : 0.0f;
    }
}

// ---------------------------------------------------------------------------
extern "C" void kernel_launch(void* const* d_in, const int* in_sizes, int n_in,
                              void* d_out, int out_size, void* d_ws, size_t ws_size,
                              hipStream_t stream) {
    (void)in_sizes; (void)n_in; (void)d_ws; (void)ws_size; (void)out_size;

    const float*         h      = (const float*)d_in[0];          // (4, 8192, 2048) f32
    const unsigned char* exited = (const unsigned char*)d_in[1];  // (4, 8192, 1) bool
    const float*         W      = (const float*)d_in[2];          // (1, 2048) f32
    const float*         bias   = (const float*)d_in[3];          // (1,) f32

    float* scores = (float*)d_out;            // first 32768 floats
    float* mask   = scores + NROWS;           // next 32768 (mask as 0/1)

    // 2048 row-tiles of 16 rows, 8 waves per block -> 256 blocks.
    gemv_sigmoid_wmma<<<dim3(NROWS / (16 * 8)), dim3(256), 0, stream>>>(h, W, bias, scores);
    topk_mask_kernel<<<dim3(B_LEN), dim3(1024), 0, stream>>>(scores, exited, mask);
}